// DCNBlock_39032662786326
// MI455X (gfx1250) — compile-verified
//
#include <hip/hip_runtime.h>

typedef float v2f __attribute__((ext_vector_type(2)));
typedef float v8f __attribute__((ext_vector_type(8)));
typedef int   v4i __attribute__((ext_vector_type(4)));

#define BB 4
#define CC 64
#define OO 64
#define HH 128
#define WW 128
#define HWW (HH * WW)
#define K2 9
#define KDIM (K2 * CC)   // 576, k index = kk*64 + c

#if defined(__AMDGCN__) && __has_builtin(__builtin_amdgcn_global_load_async_to_lds_b128) && \
    __has_builtin(__builtin_amdgcn_s_wait_asynccnt)
#define USE_ASYNC 1
#else
#define USE_ASYNC 0
#endif

#if USE_ASYNC
typedef __attribute__((address_space(1))) v4i* g4_ptr;   // global v4i*
typedef __attribute__((address_space(3))) v4i* l4_ptr;   // LDS v4i*
__device__ __forceinline__ void async_copy_b128(const void* g, void* l) {
    __builtin_amdgcn_global_load_async_to_lds_b128((g4_ptr)g, (l4_ptr)l, 0, 0);
}
__device__ __forceinline__ void async_wait0() { __builtin_amdgcn_s_wait_asynccnt(0); }
#endif

__device__ __forceinline__ v8f wmma_f32(v2f a, v2f b, v8f c) {
    // V_WMMA_F32_16X16X4_F32 : D = A(16x4) * B(4x16) + C(16x16), fp32
    return __builtin_amdgcn_wmma_f32_16x16x4_f32(false, a, false, b, (short)0, c, false, false);
}

// ---------------------------------------------------------------------------
// Kernel 1: x [B,C,H,W] -> xt [B,H,W,C]  (channel-last so corner gathers are
// 64 contiguous floats = 256B)
// ---------------------------------------------------------------------------
__global__ void dcn_transpose_nhwc(const float* __restrict__ x, float* __restrict__ xt) {
    __shared__ float tile[CC][WW + 1];
    int bh = blockIdx.x;
    int b = bh >> 7, h = bh & 127;
    int t = threadIdx.x;
    for (int c0 = 0; c0 < CC; c0 += 2) {
        int c = c0 + (t >> 7);
        int w = t & 127;
        tile[c][w] = x[(((size_t)b * CC + c) * HH + h) * WW + w];
    }
    __syncthreads();
    for (int w0 = 0; w0 < WW; w0 += 4) {
        int w = w0 + (t >> 6);
        int c = t & 63;
        xt[(((size_t)b * HH + h) * WW + w) * CC + c] = tile[c][w];
    }
}

// ---------------------------------------------------------------------------
// Kernel 2: pack weights into B-panels [k = kk*64+c][n]
//   Boff: n = 0..31 (18 valid, rest zero), Bdcn: n = 0..63
// ---------------------------------------------------------------------------
__global__ void dcn_pack_weights(const float* __restrict__ w_off, const float* __restrict__ w_dcn,
                                 float* __restrict__ Boff, float* __restrict__ Bdcn) {
    int i = blockIdx.x * blockDim.x + threadIdx.x;
    if (i < KDIM * 32) {
        int j = i & 31;
        int kc = i >> 5;              // kk*64 + c
        int kk = kc >> 6, c = kc & 63;
        Boff[i] = (j < 18) ? w_off[((size_t)j * CC + c) * K2 + kk] : 0.f;
        return;
    }
    i -= KDIM * 32;
    if (i < KDIM * 64) {
        int o = i & 63;
        int kc = i >> 6;
        int kk = kc >> 6, c = kc & 63;
        Bdcn[i] = w_dcn[((size_t)o * CC + c) * K2 + kk];
    }
}

// ---------------------------------------------------------------------------
// Kernel 3: offset conv GEMM.  One block per (b,h) row; 8 waves, each owns an
// M-tile of 16 positions.  Per kk: stage im2col A-slab (16x64) per wave and a
// shared B panel (64x32, async-to-LDS when available), then 16 K-steps x
// 2 N-tiles of f32 WMMA.  Epilogue stages through LDS so the [row][18] output
// region (contiguous 128*18 floats) is stored fully coalesced.
// ---------------------------------------------------------------------------
__global__ void dcn_offset_gemm(const float* __restrict__ xt, const float* __restrict__ Boff,
                                const float* __restrict__ boff, float* __restrict__ offs) {
    __shared__ float Aslab[8][16][68];   // row pad 68 -> conflict-free column reads
    __shared__ float Bp[CC][48];         // 48 pad -> lane halves hit disjoint banks
    __shared__ float offstile[WW][18];

    int bh = blockIdx.x;
    int b = bh >> 7, h = bh & 127;
    int wave = threadIdx.x >> 5, lane = threadIdx.x & 31;
    int w0 = wave * 16;

    v8f acc0 = {}, acc1 = {};

    for (int kk = 0; kk < K2; ++kk) {
        int ky = kk / 3, kx = kk % 3;
        // shared B panel for this kk
#if USE_ASYNC
        for (int i = threadIdx.x; i < 512; i += 256) {
            int c = i >> 3, j4 = (i & 7) << 2;
            async_copy_b128(&Boff[((size_t)(kk * CC + c)) * 32 + j4], &Bp[c][j4]);
        }
#else
        for (int i = threadIdx.x; i < 512; i += 256) {
            int c = i >> 3, j4 = (i & 7) << 2;
            *(float4*)&Bp[c][j4] = *(const float4*)&Boff[((size_t)(kk * CC + c)) * 32 + j4];
        }
#endif
        // im2col A slab: lane pair per position m, each lane covers 32 channels
        int m = lane >> 1, half = lane & 1;
        int y = h + ky - 1;
        int xw = w0 + m + kx - 1;
        bool valid = ((unsigned)y < (unsigned)HH) && ((unsigned)xw < (unsigned)WW);
        const float* src = xt + (((size_t)(b * HH + (valid ? y : 0)) * WW + (valid ? xw : 0)) * CC);
        for (int i = 0; i < 8; ++i) {
            int c = half * 32 + i * 4;
            float4 v;
            if (valid) v = *(const float4*)(src + c);
            else { v.x = v.y = v.z = v.w = 0.f; }
            *(float4*)&Aslab[wave][m][c] = v;
        }
#if USE_ASYNC
        async_wait0();
#endif
        __syncthreads();

        int ml = lane & 15;
        int klo = (lane >> 4) << 1;   // lanes 0-15: K 0,1 ; lanes 16-31: K 2,3
        const float* arow = &Aslab[wave][ml][0];
#pragma unroll 4
        for (int c0 = 0; c0 < CC; c0 += 4) {
            v2f a;  a.x = arow[c0 + klo];          a.y = arow[c0 + klo + 1];
            v2f b0; b0.x = Bp[c0 + klo][ml];       b0.y = Bp[c0 + klo + 1][ml];
            v2f b1; b1.x = Bp[c0 + klo][16 + ml];  b1.y = Bp[c0 + klo + 1][16 + ml];
            acc0 = wmma_f32(a, b0, acc0);
            acc1 = wmma_f32(a, b1, acc1);
        }
        __syncthreads();
    }

    // stage: lane = N column, vgpr r -> M = r (lanes 0-15) / r+8 (lanes 16-31)
    int n0 = lane & 15;
    int mb = (lane >> 4) * 8;
    float bj0 = boff[n0];
    float bj1 = (n0 < 2) ? boff[16 + n0] : 0.f;
#pragma unroll
    for (int r = 0; r < 8; ++r) {
        int w = w0 + mb + r;
        offstile[w][n0] = acc0[r] + bj0;
        if (n0 < 2) offstile[w][16 + n0] = acc1[r] + bj1;
    }
    __syncthreads();
    // coalesced store of the contiguous 128*18-float row region
    float* obase = offs + ((size_t)b * HWW + (size_t)h * WW) * 18;
    const float* flat = &offstile[0][0];
    for (int i = threadIdx.x; i < WW * 18; i += 256) obase[i] = flat[i];
}

// ---------------------------------------------------------------------------
// Kernel 4: deformable conv.  Same skeleton; A slab is the bilinear blend of
// 4 clamped corner float4 loads (weights zeroed when out of range), N = 64.
// Epilogue stages the 64x128 output row tile in LDS (overlaid on the A slab)
// so NCHW stores are fully coalesced per o-row.
// ---------------------------------------------------------------------------
__global__ void dcn_deform_gemm(const float* __restrict__ xt, const float* __restrict__ Bdcn,
                                const float* __restrict__ bdcn, const float* __restrict__ offs,
                                float* __restrict__ out) {
    __shared__ union SM {
        float aslab[8][16][68];   // 34,816 B
        float outt[OO][WW + 2];   // 33,280 B (pad 130 -> reduced bank conflicts)
    } sm;
    __shared__ float Bp[CC][80];  // 80 pad -> lane halves hit disjoint banks

    int bh = blockIdx.x;
    int b = bh >> 7, h = bh & 127;
    int wave = threadIdx.x >> 5, lane = threadIdx.x & 31;
    int w0 = wave * 16;

    v8f acc[4];
    acc[0] = v8f{}; acc[1] = v8f{}; acc[2] = v8f{}; acc[3] = v8f{};

    int m = lane >> 1, half = lane & 1;
    size_t pidx = (size_t)b * HWW + h * WW + w0 + m;
    const float* ob = offs + pidx * 18;
    __builtin_prefetch(ob, 0, 3);

    for (int kk = 0; kk < K2; ++kk) {
        int ky = kk / 3, kx = kk % 3;
        // shared B panel for this kk (64x64 valid floats)
#if USE_ASYNC
        for (int i = threadIdx.x; i < 1024; i += 256) {
            int c = i >> 4, o4 = (i & 15) << 2;
            async_copy_b128(&Bdcn[((size_t)(kk * CC + c)) * 64 + o4], &Bp[c][o4]);
        }
#else
        for (int i = threadIdx.x; i < 1024; i += 256) {
            int c = i >> 4, o4 = (i & 15) << 2;
            *(float4*)&Bp[c][o4] = *(const float4*)&Bdcn[((size_t)(kk * CC + c)) * 64 + o4];
        }
#endif
        // bilinear A slab
        float dy = ob[2 * kk + 0], dx = ob[2 * kk + 1];
        float py = (float)(h + ky - 1) + dy;
        float px = (float)(w0 + m + kx - 1) + dx;
        float y0f = floorf(py), x0f = floorf(px);
        int y0 = (int)y0f, x0 = (int)x0f;
        float wy = py - y0f, wx = px - x0f;
        float w00 = (1.f - wy) * (1.f - wx), w01 = (1.f - wy) * wx;
        float w10 = wy * (1.f - wx),         w11 = wy * wx;
        if ((unsigned)y0       >= (unsigned)HH) { w00 = 0.f; w01 = 0.f; }
        if ((unsigned)(y0 + 1) >= (unsigned)HH) { w10 = 0.f; w11 = 0.f; }
        if ((unsigned)x0       >= (unsigned)WW) { w00 = 0.f; w10 = 0.f; }
        if ((unsigned)(x0 + 1) >= (unsigned)WW) { w01 = 0.f; w11 = 0.f; }
        int y0c = min(max(y0, 0), HH - 1), y1c = min(max(y0 + 1, 0), HH - 1);
        int x0c = min(max(x0, 0), WW - 1), x1c = min(max(x0 + 1, 0), WW - 1);
        const float* p00 = xt + (((size_t)(b * HH + y0c) * WW + x0c) << 6);
        const float* p01 = xt + (((size_t)(b * HH + y0c) * WW + x1c) << 6);
        const float* p10 = xt + (((size_t)(b * HH + y1c) * WW + x0c) << 6);
        const float* p11 = xt + (((size_t)(b * HH + y1c) * WW + x1c) << 6);
        for (int i = 0; i < 8; ++i) {
            int c = half * 32 + i * 4;
            float4 a00 = *(const float4*)(p00 + c);
            float4 a01 = *(const float4*)(p01 + c);
            float4 a10 = *(const float4*)(p10 + c);
            float4 a11 = *(const float4*)(p11 + c);
            float4 r;
            r.x = w00 * a00.x + w01 * a01.x + w10 * a10.x + w11 * a11.x;
            r.y = w00 * a00.y + w01 * a01.y + w10 * a10.y + w11 * a11.y;
            r.z = w00 * a00.z + w01 * a01.z + w10 * a10.z + w11 * a11.z;
            r.w = w00 * a00.w + w01 * a01.w + w10 * a10.w + w11 * a11.w;
            *(float4*)&sm.aslab[wave][m][c] = r;
        }
#if USE_ASYNC
        async_wait0();
#endif
        __syncthreads();

        int ml = lane & 15;
        int klo = (lane >> 4) << 1;
        const float* arow = &sm.aslab[wave][ml][0];
#pragma unroll 2
        for (int c0 = 0; c0 < CC; c0 += 4) {
            v2f a; a.x = arow[c0 + klo]; a.y = arow[c0 + klo + 1];
#pragma unroll
            for (int nt = 0; nt < 4; ++nt) {
                v2f bb;
                bb.x = Bp[c0 + klo][nt * 16 + ml];
                bb.y = Bp[c0 + klo + 1][nt * 16 + ml];
                acc[nt] = wmma_f32(a, bb, acc[nt]);
            }
        }
        __syncthreads();
    }

    // stage the 64x128 output row tile (bias applied), overlaid on the A slab
    int n0 = lane & 15;
    int mb = (lane >> 4) * 8;
#pragma unroll
    for (int nt = 0; nt < 4; ++nt) {
        int o = nt * 16 + n0;
        float bias = bdcn[o];
#pragma unroll
        for (int r = 0; r < 8; ++r) {
            sm.outt[o][w0 + mb + r] = acc[nt][r] + bias;
        }
    }
    __syncthreads();
    // coalesced NCHW stores: one contiguous 512B row per o channel
    for (int i = threadIdx.x; i < OO * WW; i += 256) {
        int o = i >> 7, w = i & 127;
        out[(((size_t)b * OO + o) * HH + h) * WW + w] = sm.outt[o][w];
    }
}

extern "C" void kernel_launch(void* const* d_in, const int* in_sizes, int n_in,
                              void* d_out, int out_size, void* d_ws, size_t ws_size,
                              hipStream_t stream) {
    const float* x     = (const float*)d_in[0];
    const float* w_off = (const float*)d_in[1];
    const float* b_off = (const float*)d_in[2];
    const float* w_dcn = (const float*)d_in[3];
    const float* b_dcn = (const float*)d_in[4];
    float* out = (float*)d_out;

    float* ws   = (float*)d_ws;
    float* xt   = ws;                                   // 4*128*128*64      = 4,194,304 f
    float* Boff = xt + (size_t)BB * HH * WW * CC;       // 576*32            =    18,432 f
    float* Bdcn = Boff + (size_t)KDIM * 32;             // 576*64            =    36,864 f
    float* offs = Bdcn + (size_t)KDIM * 64;             // 4*16384*18        = 1,179,648 f

    dcn_transpose_nhwc<<<BB * HH, 256, 0, stream>>>(x, xt);
    dcn_pack_weights<<<(KDIM * 32 + KDIM * 64 + 255) / 256, 256, 0, stream>>>(w_off, w_dcn, Boff, Bdcn);
    dcn_offset_gemm<<<BB * HH, 256, 0, stream>>>(xt, Boff, b_off, offs);
    dcn_deform_gemm<<<BB * HH, 256, 0, stream>>>(xt, Bdcn, b_dcn, offs, out);
}